// TextDecoder_32521492365719
// MI455X (gfx1250) — compile-verified
//
#include <hip/hip_runtime.h>
#include <math.h>

#define HID   1024
#define VOCAB 50257

typedef __attribute__((ext_vector_type(2))) float v2f;
typedef __attribute__((ext_vector_type(8))) float v8f;

// monotone float -> uint key (for order-independent atomic max)
__device__ __forceinline__ unsigned f2key(float f) {
    unsigned b = __float_as_uint(f);
    return (b & 0x80000000u) ? ~b : (b | 0x80000000u);
}
__device__ __forceinline__ float key2f(unsigned k) {
    unsigned b = (k & 0x80000000u) ? (k ^ 0x80000000u) : ~k;
    return __uint_as_float(b);
}

__device__ __forceinline__ v8f wmma_f32_k4(v2f a, v2f b, v8f c) {
    // D = A(16x4) * B(4x16) + C ; f32, wave32
    return __builtin_amdgcn_wmma_f32_16x16x4_f32(
        false, a, false, b, (short)0, c, false, false);
}

// one-shot streaming read of weights: bypass temporal caching (th: NT)
__device__ __forceinline__ v2f ld_stream(const float* p) {
    return __builtin_nontemporal_load((const v2f*)p);
}

// ---------------------------------------------------------------------------
// K1: embedding+ReLU, GRU cell. 8 blocks x 256 threads; each block owns 128
// h-indices; each wave owns 16 rows of each of the 6 needed weight-row sets.
// ---------------------------------------------------------------------------
__global__ __launch_bounds__(256) void gru_kernel(
    const int* __restrict__ tok, const float* __restrict__ hidden,
    const float* __restrict__ emb,
    const float* __restrict__ w_ih, const float* __restrict__ w_hh,
    const float* __restrict__ b_ih, const float* __restrict__ b_hh,
    float* __restrict__ h_new_out, unsigned* __restrict__ gkey)
{
    __shared__ float xs[HID];
    __shared__ float hs[HID];
    __shared__ float acc[6][128];

    const int t = threadIdx.x;
    {
        const int T = tok[0];
        float4 ev = ((const float4*)(emb + (size_t)T * HID))[t];
        xs[4*t+0] = fmaxf(ev.x, 0.f);
        xs[4*t+1] = fmaxf(ev.y, 0.f);
        xs[4*t+2] = fmaxf(ev.z, 0.f);
        xs[4*t+3] = fmaxf(ev.w, 0.f);
        ((float4*)hs)[t] = ((const float4*)hidden)[t];
    }
    if (t == 0) *gkey = 0u;   // idempotent init for K2's atomic max
    __syncthreads();

    const int wave = t >> 5;
    const int lane = t & 31;
    const int m    = lane & 15;
    const int half = lane >> 4;           // 0: K=0,1   1: K=2,3
    const int j0   = blockIdx.x * 128 + wave * 16;

    const size_t row_r = (size_t)(         j0 + m) * HID;
    const size_t row_z = (size_t)(  HID  + j0 + m) * HID;
    const size_t row_n = (size_t)(2*HID  + j0 + m) * HID;
    const int kl0 = 2 * half;

    v8f c_ir = {}, c_hr = {}, c_iz = {}, c_hz = {}, c_in = {}, c_hn = {};

    #pragma unroll 2
    for (int k = 0; k < HID; k += 4) {
        const int kk = k + kl0;
        v2f bx = *(const v2f*)(xs + kk);
        v2f bh = *(const v2f*)(hs + kk);
        v2f a;
        a = ld_stream(w_ih + row_r + kk); c_ir = wmma_f32_k4(a, bx, c_ir);
        a = ld_stream(w_hh + row_r + kk); c_hr = wmma_f32_k4(a, bh, c_hr);
        a = ld_stream(w_ih + row_z + kk); c_iz = wmma_f32_k4(a, bx, c_iz);
        a = ld_stream(w_hh + row_z + kk); c_hz = wmma_f32_k4(a, bh, c_hz);
        a = ld_stream(w_ih + row_n + kk); c_in = wmma_f32_k4(a, bx, c_in);
        a = ld_stream(w_hh + row_n + kk); c_hn = wmma_f32_k4(a, bh, c_hn);
    }

    // lanes 0 / 16 hold local rows [0..7] / [8..15] (all B columns identical)
    if (m == 0) {
        const int lj = wave * 16 + half * 8;
        #pragma unroll
        for (int p = 0; p < 8; ++p) {
            acc[0][lj+p] = c_ir[p];
            acc[1][lj+p] = c_hr[p];
            acc[2][lj+p] = c_iz[p];
            acc[3][lj+p] = c_hz[p];
            acc[4][lj+p] = c_in[p];
            acc[5][lj+p] = c_hn[p];
        }
    }
    __syncthreads();

    if (t < 128) {
        const int i = blockIdx.x * 128 + t;
        float gr = acc[0][t] + b_ih[i]         + acc[1][t] + b_hh[i];
        float gz = acc[2][t] + b_ih[HID + i]   + acc[3][t] + b_hh[HID + i];
        float r  = 1.f / (1.f + expf(-gr));
        float z  = 1.f / (1.f + expf(-gz));
        float n  = tanhf(acc[4][t] + b_ih[2*HID + i] + r * (acc[5][t] + b_hh[2*HID + i]));
        float hn = (1.f - z) * n + z * hs[i];
        h_new_out[i] = hn;
    }
}

// ---------------------------------------------------------------------------
// K2: logits = w_out @ h_new + b_out  (206 MB stream — the bandwidth hog).
// 197 blocks x 256 threads; each wave owns 32 vocab rows as TWO independent
// WMMA accumulator chains (ILP; shared B chunk from LDS).
// ---------------------------------------------------------------------------
__global__ __launch_bounds__(256) void logits_kernel(
    const float* __restrict__ w_out, const float* __restrict__ b_out,
    const float* __restrict__ h_new, float* __restrict__ logits,
    unsigned* __restrict__ gkey)
{
    __shared__ float hs[HID];
    __shared__ float red[256];

    const int t = threadIdx.x;
    ((float4*)hs)[t] = ((const float4*)h_new)[t];
    __syncthreads();

    const int wave = t >> 5;
    const int lane = t & 31;
    const int m    = lane & 15;
    const int half = lane >> 4;
    const long r0  = (long)blockIdx.x * 256 + wave * 32;
    long rowA = r0 + m;      if (rowA > VOCAB - 1) rowA = VOCAB - 1;  // tail clamp
    long rowB = r0 + 16 + m; if (rowB > VOCAB - 1) rowB = VOCAB - 1;
    const float* wrowA = w_out + rowA * (long)HID;
    const float* wrowB = w_out + rowB * (long)HID;
    const int kl0 = 2 * half;

    v8f c0 = {}, c1 = {};
    #pragma unroll 4
    for (int k = 0; k < HID; k += 4) {
        const int kk = k + kl0;
        v2f b  = *(const v2f*)(hs + kk);
        v2f a0 = ld_stream(wrowA + kk);
        v2f a1 = ld_stream(wrowB + kk);
        c0 = wmma_f32_k4(a0, b, c0);
        c1 = wmma_f32_k4(a1, b, c1);
    }

    float lmax = -INFINITY;
    if (m == 0) {
        const long base0 = r0 + half * 8;
        const long base1 = r0 + 16 + half * 8;
        #pragma unroll
        for (int p = 0; p < 8; ++p) {
            long row = base0 + p;
            if (row < VOCAB) {
                float v = c0[p] + b_out[row];
                logits[row] = v;
                lmax = fmaxf(lmax, v);
            }
            row = base1 + p;
            if (row < VOCAB) {
                float v = c1[p] + b_out[row];
                logits[row] = v;
                lmax = fmaxf(lmax, v);
            }
        }
    }

    red[t] = lmax;
    __syncthreads();
    for (int s = 128; s > 0; s >>= 1) {
        if (t < s) red[t] = fmaxf(red[t], red[t + s]);
        __syncthreads();
    }
    if (t == 0) atomicMax(gkey, f2key(red[0]));
}

// ---------------------------------------------------------------------------
// K3: logZ = max + log(sum(exp(logit - max))); single block, fixed-tree reduce
// ---------------------------------------------------------------------------
__global__ __launch_bounds__(1024) void sumexp_kernel(
    const float* __restrict__ logits, const unsigned* __restrict__ gkey,
    float* __restrict__ logz)
{
    __shared__ float red[1024];
    const int t = threadIdx.x;
    const float gmax = key2f(*gkey);
    float s = 0.f;
    for (int i = t; i < VOCAB; i += 1024) s += expf(logits[i] - gmax);
    red[t] = s;
    __syncthreads();
    for (int st = 512; st > 0; st >>= 1) {
        if (t < st) red[t] += red[t + st];
        __syncthreads();
    }
    if (t == 0) *logz = gmax + logf(red[0]);
}

// ---------------------------------------------------------------------------
// K4: out[i] = logits[i] - logZ
// ---------------------------------------------------------------------------
__global__ __launch_bounds__(256) void sub_kernel(
    float* __restrict__ out, const float* __restrict__ logz)
{
    const int i = blockIdx.x * 256 + threadIdx.x;
    if (i < VOCAB) out[i] -= *logz;
}

extern "C" void kernel_launch(void* const* d_in, const int* in_sizes, int n_in,
                              void* d_out, int out_size, void* d_ws, size_t ws_size,
                              hipStream_t stream) {
    const int*   tok    = (const int*)  d_in[0];
    const float* hidden = (const float*)d_in[1];
    const float* emb    = (const float*)d_in[2];
    const float* w_ih   = (const float*)d_in[3];
    const float* w_hh   = (const float*)d_in[4];
    const float* b_ih   = (const float*)d_in[5];
    const float* b_hh   = (const float*)d_in[6];
    const float* w_out  = (const float*)d_in[7];
    const float* b_out  = (const float*)d_in[8];

    float*    out  = (float*)d_out;          // [0,V): log-softmax, [V,V+H): h_new
    unsigned* gkey = (unsigned*)d_ws;        // 1 uint  (max key)
    float*    logz = (float*)d_ws + 1;       // 1 float (logZ)

    gru_kernel   <<<8,                    256,  0, stream>>>(tok, hidden, emb, w_ih, w_hh,
                                                             b_ih, b_hh, out + VOCAB, gkey);
    logits_kernel<<<(VOCAB + 255) / 256,  256,  0, stream>>>(w_out, b_out, out + VOCAB, out, gkey);
    sumexp_kernel<<<1,                    1024, 0, stream>>>(out, gkey, logz);
    sub_kernel   <<<(VOCAB + 255) / 256,  256,  0, stream>>>(out, logz);
}